// TimeAwareAttentionModel_4329327035192
// MI455X (gfx1250) — compile-verified
//
#include <hip/hip_runtime.h>
#include <hip/hip_bf16.h>

// Problem constants (compile-time, from the reference)
#define Nn   10000
#define Ee   160000
#define DE   64
#define HIDN 128
#define NODE_ELEMS 1024            // C*H*W = 16*64
#define OUT0 (Nn * NODE_ELEMS)     // offset of dec_edge_feats in d_out

typedef __attribute__((ext_vector_type(2))) float v2f;
typedef __attribute__((ext_vector_type(8))) float v8f;

#if defined(__has_builtin)
#if __has_builtin(__builtin_amdgcn_global_load_async_to_lds_b128)
#define USE_ASYNC_LDS 1
#endif
#endif

#if USE_ASYNC_LDS
// Builtin prototype (from hipcc diagnostic): param0 = int __vector(4) AS1*,
// param1 = LDS-side pointer (AS3), then imm offset / imm cpol.
typedef int v4i __attribute__((vector_size(16)));
typedef __attribute__((address_space(1))) v4i* gv4p;
typedef __attribute__((address_space(3))) v4i* lv4p;
#endif

__device__ __forceinline__ v8f wmma4(v2f a, v2f b, v8f c) {
    // D = A(16x4,f32) x B(4x16,f32) + C(16x16,f32), exact f32 path on CDNA5
    return __builtin_amdgcn_wmma_f32_16x16x4_f32(false, a, false, b,
                                                 (short)0, c, false, false);
}

__device__ __forceinline__ void atomicMaxFloat(float* addr, float value) {
    if (value >= 0.0f)
        atomicMax((int*)addr, __float_as_int(value));
    else
        atomicMin((unsigned int*)addr, __float_as_uint(value));
}

// -------- Kernel 1: edge MLP -> logits (written to d_out tail) -------------
// One wave computes 16 edges; 8 hidden tiles x 16 k-chunks of V_WMMA_F32_16X16X4_F32.
__global__ __launch_bounds__(256) void k_edge_mlp(
    const float* __restrict__ ea, const float* __restrict__ W1,
    const float* __restrict__ b1, const float* __restrict__ W2,
    const float* __restrict__ b2, float* __restrict__ out)
{
    __shared__ float w1s[DE * HIDN];          // 32 KB staged B operand
#if USE_ASYNC_LDS
    {
        // Async global->LDS: 2048 float4; 256 threads x 8 iters x 16B.
        // Each lane supplies (global addr, LDS addr); ASYNCcnt tracks completion.
        const float4* g = (const float4*)W1;
#pragma unroll
        for (int it = 0; it < 8; ++it) {
            int i = it * 256 + threadIdx.x;   // float4 index
            __builtin_amdgcn_global_load_async_to_lds_b128(
                (gv4p)(float4*)(g + i),
                (lv4p)(w1s + 4 * i),
                0, 0);
        }
#if __has_builtin(__builtin_amdgcn_s_wait_asynccnt)
        __builtin_amdgcn_s_wait_asynccnt(0);
#else
        asm volatile("s_wait_asynccnt 0x0" ::: "memory");
#endif
    }
#else
    for (int i = threadIdx.x; i < DE * HIDN; i += 256) w1s[i] = W1[i];
#endif
    __syncthreads();

    const int lane = threadIdx.x & 31;
    const int wid  = threadIdx.x >> 5;
    const int half = lane >> 4;               // 0: K pair {0,1}, 1: {2,3}
    const int ll   = lane & 15;
    const int koff = half ? 2 : 0;
    const int ebase = blockIdx.x * 128 + wid * 16;

    // Preload A (this wave's 16x64 edge_attr tile), 2 floats/lane/chunk
    const float* ar = ea + (size_t)(ebase + ll) * DE;
    v2f a[16];
#pragma unroll
    for (int k = 0; k < 16; ++k) {
        a[k].x = ar[4 * k + koff];
        a[k].y = ar[4 * k + koff + 1];
    }

    float S[8];
#pragma unroll
    for (int r = 0; r < 8; ++r) S[r] = 0.0f;

#pragma unroll
    for (int j = 0; j < 8; ++j) {             // hidden tiles (128 = 8*16)
        const int col = 16 * j + ll;
        const float b1c = b1[col];
        v8f acc;
#pragma unroll
        for (int r = 0; r < 8; ++r) acc[r] = b1c;   // fused bias
#pragma unroll
        for (int k = 0; k < 16; ++k) {        // K = 64 = 16*4
            v2f b;
            b.x = w1s[(4 * k + koff) * HIDN + col];
            b.y = w1s[(4 * k + koff + 1) * HIDN + col];
            acc = wmma4(a[k], b, acc);
        }
        const float w2v = W2[col];            // second layer, fused relu+dot
#pragma unroll
        for (int r = 0; r < 8; ++r) S[r] += fmaxf(acc[r], 0.0f) * w2v;
    }

    // butterfly sum across the 16 lanes of each half (cols of the tile)
#pragma unroll
    for (int m = 1; m < 16; m <<= 1) {
#pragma unroll
        for (int r = 0; r < 8; ++r) S[r] += __shfl_xor(S[r], m, 32);
    }
    const float b2v = b2[0];
    if (ll == 0) {                            // lane0 -> edges M=0..7, lane16 -> M=8..15
#pragma unroll
        for (int r = 0; r < 8; ++r)
            out[OUT0 + ebase + r + 8 * half] = S[r] + b2v;
    }
}

// -------- Kernel 2: per-node precompute: base conv term + y_in/y_out -------
// out_base[n] = Wn[:,0:16] (x) x[n] + bn ; y_in = Wn[:,16:32] (x) x[n] ; y_out = Wn[:,32:48] (x) x[n]
// Each is a 16x16x(K=16) GEMM over the 64 spatial positions -> WMMA f32 16x16x4.
__global__ __launch_bounds__(256) void k_node_pre(
    const float* __restrict__ x, const float* __restrict__ Wn,
    const float* __restrict__ bn, float* __restrict__ out,
    float* __restrict__ yin, float* __restrict__ yout)
{
    const int lane = threadIdx.x & 31;
    const int wid  = threadIdx.x >> 5;
    const int half = lane >> 4;
    const int ll   = lane & 15;
    const int koff = half ? 2 : 0;
    const int n = blockIdx.x * 8 + wid;       // one wave per node

    // A operands: Wn[o][c], o = ll (M), c = 16*m + 4*k + koff{,+1}
    v2f awn[3][4];
#pragma unroll
    for (int m = 0; m < 3; ++m)
#pragma unroll
        for (int k = 0; k < 4; ++k) {
            awn[m][k].x = Wn[ll * 48 + 16 * m + 4 * k + koff];
            awn[m][k].y = Wn[ll * 48 + 16 * m + 4 * k + koff + 1];
        }
    float bnv[8];
#pragma unroll
    for (int r = 0; r < 8; ++r) bnv[r] = bn[r + 8 * half];

    const float* xr = x + (size_t)n * NODE_ELEMS;
#pragma unroll
    for (int j = 0; j < 4; ++j) {             // spatial tiles (64 = 4*16)
        v8f aB, aI, aO;
#pragma unroll
        for (int r = 0; r < 8; ++r) { aB[r] = bnv[r]; aI[r] = 0.0f; aO[r] = 0.0f; }
#pragma unroll
        for (int k = 0; k < 4; ++k) {         // K = 16 channels = 4*4
            v2f b;                            // B = x[n]: rows c, cols spatial
            b.x = xr[(4 * k + koff) * 64 + 16 * j + ll];
            b.y = xr[(4 * k + koff + 1) * 64 + 16 * j + ll];
            aB = wmma4(awn[0][k], b, aB);     // same B reused 3x
            aI = wmma4(awn[1][k], b, aI);
            aO = wmma4(awn[2][k], b, aO);
        }
        const int p = 16 * j + ll;
#pragma unroll
        for (int r = 0; r < 8; ++r) {
            const size_t o = (size_t)n * NODE_ELEMS + (r + 8 * half) * 64 + p;
            out[o]  = aB[r];
            yin[o]  = aI[r];
            yout[o] = aO[r];
        }
    }
}

// -------- Kernel 3: init softmax stats ------------------------------------
__global__ void k_init_stats(float* m_out, float* m_in, float* s_out, float* s_in) {
    int i = blockIdx.x * 256 + threadIdx.x;
    if (i < Nn) {
        m_out[i] = -__builtin_inff();
        m_in[i]  = -__builtin_inff();
        s_out[i] = 0.0f;
        s_in[i]  = 0.0f;
    }
}

// -------- Kernel 4: segment max (masked) ----------------------------------
__global__ void k_seg_max(const int* __restrict__ idx, const float* __restrict__ out,
                          float* __restrict__ m_out, float* __restrict__ m_in) {
    int e = blockIdx.x * 256 + threadIdx.x;
    int row = idx[e], col = idx[Ee + e];
    float lg = out[OUT0 + e];
    if (row < col)      atomicMaxFloat(&m_out[row], lg);
    else if (row > col) atomicMaxFloat(&m_in[row], lg);
}

// -------- Kernel 5: segment sum of exp ------------------------------------
__global__ void k_seg_sum(const int* __restrict__ idx, const float* __restrict__ out,
                          const float* __restrict__ m_out, const float* __restrict__ m_in,
                          float* __restrict__ s_out, float* __restrict__ s_in) {
    int e = blockIdx.x * 256 + threadIdx.x;
    int row = idx[e], col = idx[Ee + e];
    float lg = out[OUT0 + e];
    if (row < col)      atomicAdd(&s_out[row], expf(lg - m_out[row]));
    else if (row > col) atomicAdd(&s_in[row],  expf(lg - m_in[row]));
}

// -------- Kernel 6: weighted scatter of premultiplied messages ------------
// One wave per edge: d_out[row] += w_e * y_flow[col]  (1024 floats)
__global__ __launch_bounds__(256) void k_scatter(
    const int* __restrict__ idx,
    const float* __restrict__ outL,
    const float* __restrict__ m_out, const float* __restrict__ m_in,
    const float* __restrict__ s_out, const float* __restrict__ s_in,
    const float* __restrict__ yin, const float* __restrict__ yout,
    float* __restrict__ out)
{
    const int lane = threadIdx.x & 31;
    const int wid  = threadIdx.x >> 5;
    const int e = blockIdx.x * 8 + wid;
    const int row = idx[e], col = idx[Ee + e];
    if (row == col) return;                   // masked out of both flows

    const float lg = outL[OUT0 + e];
    float m, s;
    const float* src;
    if (row < col) { m = m_out[row]; s = s_out[row]; src = yout; }   // flow_out
    else           { m = m_in[row];  s = s_in[row];  src = yin;  }   // flow_in
    const float w = expf(lg - m) / fmaxf(s, 1e-30f);

    const float4* sp = (const float4*)(src + (size_t)col * NODE_ELEMS);
    float* dp = out + (size_t)row * NODE_ELEMS;
#pragma unroll
    for (int it = 0; it < 8; ++it) {          // 256 float4 = 1024 floats
        int q = it * 32 + lane;
        float4 v = sp[q];
        atomicAdd(dp + 4 * q + 0, w * v.x);
        atomicAdd(dp + 4 * q + 1, w * v.y);
        atomicAdd(dp + 4 * q + 2, w * v.z);
        atomicAdd(dp + 4 * q + 3, w * v.w);
    }
}

extern "C" void kernel_launch(void* const* d_in, const int* in_sizes, int n_in,
                              void* d_out, int out_size, void* d_ws, size_t ws_size,
                              hipStream_t stream) {
    const float* x  = (const float*)d_in[0];
    const float* ea = (const float*)d_in[1];
    const float* W1 = (const float*)d_in[2];
    const float* b1 = (const float*)d_in[3];
    const float* W2 = (const float*)d_in[4];
    const float* b2 = (const float*)d_in[5];
    const float* Wn = (const float*)d_in[6];
    const float* bn = (const float*)d_in[7];
    const int*  idx = (const int*)d_in[8];
    float* out = (float*)d_out;

    float* ws    = (float*)d_ws;
    float* yin   = ws;                                  // N*1024
    float* yout  = yin  + (size_t)Nn * NODE_ELEMS;      // N*1024
    float* m_out = yout + (size_t)Nn * NODE_ELEMS;      // N
    float* m_in  = m_out + Nn;
    float* s_out = m_in  + Nn;
    float* s_in  = s_out + Nn;

    // E = 160000 = 128*1250 edges; N = 10000 = 8*1250 nodes (exact tiles)
    k_edge_mlp <<<1250, 256, 0, stream>>>(ea, W1, b1, W2, b2, out);
    k_init_stats<<<(Nn + 255) / 256, 256, 0, stream>>>(m_out, m_in, s_out, s_in);
    k_node_pre <<<1250, 256, 0, stream>>>(x, Wn, bn, out, yin, yout);
    k_seg_max  <<<Ee / 256, 256, 0, stream>>>(idx, out, m_out, m_in);
    k_seg_sum  <<<Ee / 256, 256, 0, stream>>>(idx, out, m_out, m_in, s_out, s_in);
    k_scatter  <<<Ee / 8, 256, 0, stream>>>(idx, out, m_out, m_in, s_out, s_in,
                                            yin, yout, out);
}